// MoleculeConv_85873576116227
// MI455X (gfx1250) — compile-verified
//
#include <hip/hip_runtime.h>
#include <math.h>

typedef __attribute__((ext_vector_type(2))) float v2f;
typedef __attribute__((ext_vector_type(8))) float v8f;

#define DEPTH 3
#define INNER 32
#define UNITS 512
#define NNODE 100
#define NB    128
#define MPAD  112          // 7 tiles of 16 rows
#define ALD   33           // LDS row stride for attr/pre (gcd(33,64)=1 -> conflict free)
#define LLD   516          // LDS row stride for logits (516%64=4)

// ---- workspace layout (float elements) ----
#define WS_PRE    0
#define WS_ATTR   (NB*NNODE*INNER)                       // 409600
#define WS_BONDS  (2*NB*NNODE*INNER)                     // 819200
#define WS_ROWSUM (2*NB*NNODE*INNER + NB*NNODE*NNODE)    // 2099200
// total 2112000 floats = 8.45 MB

// ---------------------------------------------------------------------------
// Phase 1: one block per (b,i). Streams the contiguous 100x33 slab x[b,i,:,:]
// once (fully coalesced: 264 threads = 8 rows x 33 feats per step) producing
//   pre[b,i,f]  = sum_j x[b,i,j,f]      (f<32)
//   attr0[b,i,f]= x[b,i,i,f]            (f<32)
//   bonds[b,i,j]= x[b,i,j,32]
// This is the only pass that touches the 169 MB input -> HBM-bound phase.
// ---------------------------------------------------------------------------
__global__ __launch_bounds__(264) void mc_phase1(const float* __restrict__ x,
                                                 float* __restrict__ ws) {
  int blk = blockIdx.x;
  int b = blk / NNODE, i = blk - b * NNODE;
  const float* row = x + (size_t)(b * NNODE + i) * NNODE * 33;
  int t  = threadIdx.x;
  int f  = t % 33, jj = t / 33;
  float* bonds = ws + WS_BONDS + (size_t)(b * NNODE + i) * NNODE;

  float acc = 0.f;
  for (int it = 0; it < 13; ++it) {
    int j = jj + 8 * it;
    if (j < NNODE) {
      float v = row[(size_t)j * 33 + f];   // addr = base + 264*it + t  (coalesced)
      acc += v;
      if (f == 32) bonds[j] = v;
    }
  }
  __shared__ float s_red[8 * 33];
  s_red[jj * 33 + f] = acc;
  __syncthreads();
  if (jj == 0 && f < 32) {
    float s = 0.f;
    for (int r = 0; r < 8; ++r) s += s_red[r * 33 + f];
    ws[WS_PRE + (size_t)(b * NNODE + i) * INNER + f] = s;
  }
  if (t < 32)
    ws[WS_ATTR + (size_t)(b * NNODE + i) * INNER + t] = row[(size_t)i * 33 + t];
}

// ---------------------------------------------------------------------------
// Phase 2: rowsum_boa[b,c] = 1 + sum_r bonds[b,r,c]  (5 MB, L2-resident,
// consecutive threads read consecutive c -> coalesced)
// ---------------------------------------------------------------------------
__global__ void mc_phase2(float* __restrict__ ws) {
  int b = blockIdx.x;
  int c = threadIdx.x;
  if (c < NNODE) {
    const float* bonds = ws + WS_BONDS + (size_t)b * NNODE * NNODE;
    float s = 0.f;
    for (int r = 0; r < NNODE; ++r) s += bonds[r * NNODE + c];
    ws[WS_ROWSUM + b * NNODE + c] = 1.f + s;
  }
}

// ---- WMMA f32 16x16x4 fragment loaders ------------------------------------
// A (16x4): VGPR0 = K {0 | 2}, VGPR1 = K {1 | 3} for lanes {0-15 | 16-31}, M = lane&15
__device__ __forceinline__ v2f ldA_lds(const float* s, int m0, int k0, int lr, int half) {
  const float* p = s + (m0 + lr) * ALD + k0 + 2 * half;
  v2f a; a.x = p[0]; a.y = p[1];
  return a;
}
// B (4x16), mirrored layout: rows K {0,1 | 2,3}, N = n0 + (lane&15)
__device__ __forceinline__ v2f ldB_glb(const float* __restrict__ W, int ldw,
                                       int k0, int n0, int lr, int half) {
  v2f bb;
  bb.x = W[(k0 + 2 * half)     * ldw + n0 + lr];
  bb.y = W[(k0 + 2 * half + 1) * ldw + n0 + lr];
  return bb;
}

// ---------------------------------------------------------------------------
// Phase 3: one block (256 thr = 8 wave32) per graph. State (100x32 attr/pre,
// rowsum) lives in LDS. All GEMMs via V_WMMA_F32_16X16X4_F32. Softmax per
// 16-row tile; each wave owns a 64-column fp slice in registers (no atomics,
// deterministic row order == reference sum(axis=0)).
// ---------------------------------------------------------------------------
__global__ __launch_bounds__(256) void mc_phase3(const float* __restrict__ Wi,
                                                 const float* __restrict__ bi,
                                                 const float* __restrict__ Wo,
                                                 const float* __restrict__ bo,
                                                 const float* __restrict__ ws,
                                                 float* __restrict__ out) {
  __shared__ float s_attr[MPAD * ALD];
  __shared__ float s_pre [MPAD * ALD];
  __shared__ float s_log [16 * LLD];
  __shared__ float s_rowsum[MPAD];
  __shared__ float s_rmax[16];
  __shared__ float s_rden[16];

  int b    = blockIdx.x;
  int t    = threadIdx.x;
  int lane = t & 31, wave = t >> 5;
  int lr   = lane & 15, half = lane >> 4;

  const float* pre_g  = ws + WS_PRE    + (size_t)b * NNODE * INNER;
  const float* attr_g = ws + WS_ATTR   + (size_t)b * NNODE * INNER;
  const float* rs_g   = ws + WS_ROWSUM + b * NNODE;

  for (int idx = t; idx < MPAD * INNER; idx += 256) {
    int m = idx >> 5, f = idx & 31;
    float a0 = 0.f, p0 = 0.f;
    if (m < NNODE) { a0 = attr_g[m * INNER + f]; p0 = pre_g[m * INNER + f]; }
    s_attr[m * ALD + f] = a0;
    s_pre [m * ALD + f] = p0;
  }
  for (int m = t; m < MPAD; m += 256) s_rowsum[m] = (m < NNODE) ? rs_g[m] : 0.f;
  __syncthreads();

  float fp0 = 0.f, fp1 = 0.f;
  int c0 = wave * 64 + lane, c1 = c0 + 32;   // fp columns owned by this lane

  for (int d = 0; d <= DEPTH; ++d) {
    if (d > 0) {
      // core = pre[:,:32] @ W_inner[d] + b_inner[d]; update attr & pre in place
      const float* W  = Wi + d * INNER * INNER;
      const float* bv = bi + d * INNER;
      for (int tt = wave; tt < 14; tt += 8) {
        int mt = tt % 7, nt = tt / 7;
        v8f c = {};
        for (int k0 = 0; k0 < INNER; k0 += 4) {
          v2f a  = ldA_lds(s_pre, mt * 16, k0, lr, half);
          v2f bb = ldB_glb(W, INNER, k0, nt * 16, lr, half);
          c = __builtin_amdgcn_wmma_f32_16x16x4_f32(false, a, false, bb,
                                                    (short)0, c, false, false);
        }
        int n = nt * 16 + lr;
        float bias = bv[n];
        for (int v = 0; v < 8; ++v) {
          int m = mt * 16 + v + 8 * half;
          float core  = c[v] + bias;
          float old   = s_attr[m * ALD + n];
          float delta = core - old;
          s_attr[m * ALD + n] = core;
          s_pre [m * ALD + n] += s_rowsum[m] * delta;   // pre_{d+1} closed form
        }
      }
      __syncthreads();
    }
    // fp += softmax(attr[:,:32] @ W_output[d] + b_output[d]).sum(axis=0)
    const float* W  = Wo + d * INNER * UNITS;
    const float* bv = bo + d * UNITS;
    for (int mt = 0; mt < 7; ++mt) {
      for (int q = 0; q < 4; ++q) {              // wave covers 4 col-tiles = 64 cols
        int n0 = (wave * 4 + q) * 16;
        v8f c = {};
        for (int k0 = 0; k0 < INNER; k0 += 4) {
          v2f a  = ldA_lds(s_attr, mt * 16, k0, lr, half);
          v2f bb = ldB_glb(W, UNITS, k0, n0, lr, half);
          c = __builtin_amdgcn_wmma_f32_16x16x4_f32(false, a, false, bb,
                                                    (short)0, c, false, false);
        }
        int n = n0 + lr;
        float bias = bv[n];
        for (int v = 0; v < 8; ++v) s_log[(v + 8 * half) * LLD + n] = c[v] + bias;
      }
      __syncthreads();
      // per-row max + denominator (2 rows per wave, wave32 shuffle reductions)
      for (int rr = 0; rr < 2; ++rr) {
        int r = wave * 2 + rr;
        if (mt * 16 + r < NNODE) {
          const float* Lr = s_log + r * LLD;
          float mx = -3.4e38f;
          for (int cc = lane; cc < UNITS; cc += 32) mx = fmaxf(mx, Lr[cc]);
          for (int off = 16; off > 0; off >>= 1) mx = fmaxf(mx, __shfl_xor(mx, off, 32));
          float sm = 0.f;
          for (int cc = lane; cc < UNITS; cc += 32) sm += __expf(Lr[cc] - mx);
          for (int off = 16; off > 0; off >>= 1) sm += __shfl_xor(sm, off, 32);
          if (lane == 0) { s_rmax[r] = mx; s_rden[r] = sm; }
        }
      }
      __syncthreads();
      int nv = NNODE - mt * 16; if (nv > 16) nv = 16;
      for (int r = 0; r < nv; ++r) {
        float mx = s_rmax[r], dn = s_rden[r];
        fp0 += __expf(s_log[r * LLD + c0] - mx) / dn;
        fp1 += __expf(s_log[r * LLD + c1] - mx) / dn;
      }
      __syncthreads();
    }
  }
  out[b * UNITS + c0] = fp0;
  out[b * UNITS + c1] = fp1;
}

extern "C" void kernel_launch(void* const* d_in, const int* in_sizes, int n_in,
                              void* d_out, int out_size, void* d_ws, size_t ws_size,
                              hipStream_t stream) {
  (void)in_sizes; (void)n_in; (void)out_size; (void)ws_size;
  const float* x  = (const float*)d_in[0];
  const float* Wi = (const float*)d_in[1];
  const float* bi = (const float*)d_in[2];
  const float* Wo = (const float*)d_in[3];
  const float* bo = (const float*)d_in[4];
  float* ws  = (float*)d_ws;
  float* out = (float*)d_out;

  mc_phase1<<<dim3(NB * NNODE), dim3(264), 0, stream>>>(x, ws);
  mc_phase2<<<dim3(NB), dim3(128), 0, stream>>>(ws);
  mc_phase3<<<dim3(NB), dim3(256), 0, stream>>>(Wi, bi, Wo, bo, ws, out);
}